// TypeSpecificGATLayer_36885179138295
// MI455X (gfx1250) — compile-verified
//
#include <hip/hip_runtime.h>
#include <hip/hip_bf16.h>
#include <math.h>

typedef __attribute__((ext_vector_type(2))) float v2f;
typedef __attribute__((ext_vector_type(8))) float v8f;

#define D 128
#define SLOPE 0.2f

// ---------------------------------------------------------------------------
// out[n, d] = bias[d]   (out viewed as float4: 32 float4 per row)
// ---------------------------------------------------------------------------
__global__ void init_out_kernel(float* __restrict__ out,
                                const float* __restrict__ bias, long total4) {
    long i = (long)blockIdx.x * blockDim.x + threadIdx.x;
    if (i < total4) {
        ((float4*)out)[i] = ((const float4*)bias)[i & 31];
    }
}

// ---------------------------------------------------------------------------
// e_max = -inf, e_sum = 0
// ---------------------------------------------------------------------------
__global__ void clear_kernel(float* __restrict__ e_max,
                             float* __restrict__ e_sum, int n) {
    int i = blockIdx.x * blockDim.x + threadIdx.x;
    if (i < n) {
        e_max[i] = -__builtin_inff();
        e_sum[i] = 0.0f;
    }
}

// ---------------------------------------------------------------------------
// Wh = H @ W  via v_wmma_f32_16x16x4_f32.
// Block = 128 threads = 4 waves; each wave computes 16 rows x 128 cols.
// W (128x128 fp32, 64 KB) staged in LDS once per block.
// ---------------------------------------------------------------------------
__global__ __launch_bounds__(128)
void gat_gemm_kernel(const float* __restrict__ H, const float* __restrict__ W,
                     float* __restrict__ Wh, int nNodes) {
    __shared__ float lW[D * D];  // 64 KB

    // Cooperative load of W into LDS (float4: 4096 vectors / 128 threads).
    {
        const float4* Wv = (const float4*)W;
        float4* lWv = (float4*)lW;
        for (int i = threadIdx.x; i < (D * D) / 4; i += 128) lWv[i] = Wv[i];
    }
    __syncthreads();

    const int wave  = threadIdx.x >> 5;
    const int lane  = threadIdx.x & 31;
    const int mlane = lane & 15;
    const int khalf = lane >> 4;            // 0: K pair {0,1}, 1: K pair {2,3}
    const int row0  = (blockIdx.x * 4 + wave) * 16;

    // A fragment source row (clamped; bogus rows are never stored).
    int arow = row0 + mlane;
    if (arow >= nNodes) arow = nNodes - 1;
    const float* Arow = H + (size_t)arow * D;

    v8f acc[8] = {};

    for (int k = 0; k < D; k += 4) {
        v2f a;
        a.x = Arow[k + khalf * 2 + 0];
        a.y = Arow[k + khalf * 2 + 1];
#pragma unroll
        for (int t = 0; t < 8; ++t) {
            const int n = t * 16 + mlane;
            v2f b;
            b.x = lW[(k + khalf * 2 + 0) * D + n];
            b.y = lW[(k + khalf * 2 + 1) * D + n];
            acc[t] = __builtin_amdgcn_wmma_f32_16x16x4_f32(
                false, a, false, b, (short)0, acc[t], false, false);
        }
    }

    // D layout: acc[t][v] = Wh[row0 + v + 8*khalf][t*16 + mlane]
#pragma unroll
    for (int v = 0; v < 8; ++v) {
        const int m = row0 + v + 8 * khalf;
        if (m < nNodes) {
            float* dst = Wh + (size_t)m * D;
#pragma unroll
            for (int t = 0; t < 8; ++t) dst[t * 16 + mlane] = acc[t][v];
        }
    }
}

// ---------------------------------------------------------------------------
// s_dst[i] = Wh[i,:] . a[0:128] ; s_src[i] = Wh[i,:] . a[128:256]
// One wave per node row.
// ---------------------------------------------------------------------------
__global__ __launch_bounds__(256)
void score_kernel(const float* __restrict__ Wh, const float* __restrict__ a,
                  float* __restrict__ s_dst, float* __restrict__ s_src, int nNodes) {
    const int wave = threadIdx.x >> 5;
    const int lane = threadIdx.x & 31;
    const int rowi = blockIdx.x * 8 + wave;
    if (rowi >= nNodes) return;

    float4 x  = ((const float4*)(Wh + (size_t)rowi * D))[lane];
    float4 d4 = ((const float4*)a)[lane];
    float4 s4 = ((const float4*)(a + D))[lane];

    float pd = x.x * d4.x + x.y * d4.y + x.z * d4.z + x.w * d4.w;
    float ps = x.x * s4.x + x.y * s4.y + x.z * s4.z + x.w * s4.w;
#pragma unroll
    for (int off = 16; off > 0; off >>= 1) {
        pd += __shfl_xor(pd, off, 32);
        ps += __shfl_xor(ps, off, 32);
    }
    if (lane == 0) {
        s_dst[rowi] = pd;
        s_src[rowi] = ps;
    }
}

// ---------------------------------------------------------------------------
// e = leaky_relu(s_dst[row] + s_src[col]); atomic float-max into e_max[row].
// Float max via signed-int-max (v>=0) / unsigned-min (v<0) bit trick.
// ---------------------------------------------------------------------------
__global__ __launch_bounds__(256)
void edge_max_kernel(const int* __restrict__ row, const int* __restrict__ col,
                     const float* __restrict__ s_dst, const float* __restrict__ s_src,
                     float* __restrict__ e_buf, float* __restrict__ e_max, int E) {
    int e = blockIdx.x * blockDim.x + threadIdx.x;
    if (e >= E) return;
    const int r = row[e], c = col[e];
    float v = s_dst[r] + s_src[c];
    v = (v > 0.0f) ? v : SLOPE * v;
    e_buf[e] = v;
    if (v >= 0.0f)
        atomicMax((int*)(e_max + r), __float_as_int(v));
    else
        atomicMin((unsigned int*)(e_max + r), __float_as_uint(v));
}

// ---------------------------------------------------------------------------
// e_shift = exp(e - e_max[row]); atomicAdd into e_sum[row]; e_buf overwritten.
// ---------------------------------------------------------------------------
__global__ __launch_bounds__(256)
void edge_exp_kernel(const int* __restrict__ row, float* __restrict__ e_buf,
                     const float* __restrict__ e_max, float* __restrict__ e_sum, int E) {
    int e = blockIdx.x * blockDim.x + threadIdx.x;
    if (e >= E) return;
    const int r = row[e];
    float sh = __expf(e_buf[e] - e_max[r]);
    e_buf[e] = sh;
    unsafeAtomicAdd(e_sum + r, sh);  // global_atomic_add_f32
}

// ---------------------------------------------------------------------------
// out[row] += (e_shift / max(e_sum[row],1e-12)) * Wh[col]
// One wave per edge; each lane handles 4 contiguous floats.
// ---------------------------------------------------------------------------
__global__ __launch_bounds__(256)
void agg_kernel(const int* __restrict__ row, const int* __restrict__ col,
                const float* __restrict__ e_buf, const float* __restrict__ e_sum,
                const float* __restrict__ Wh, float* __restrict__ out, int E) {
    const int wave = threadIdx.x >> 5;
    const int lane = threadIdx.x & 31;
    const int e = blockIdx.x * 8 + wave;
    if (e >= E) return;

    const int r = row[e], c = col[e];
    float denom = e_sum[r];
    denom = (denom > 1e-12f) ? denom : 1e-12f;
    const float alpha = e_buf[e] / denom;

    float4 x = ((const float4*)(Wh + (size_t)c * D))[lane];
    float* dst = out + (size_t)r * D + lane * 4;
    unsafeAtomicAdd(dst + 0, alpha * x.x);
    unsafeAtomicAdd(dst + 1, alpha * x.y);
    unsafeAtomicAdd(dst + 2, alpha * x.z);
    unsafeAtomicAdd(dst + 3, alpha * x.w);
}

// ---------------------------------------------------------------------------
// Input order: 0:H, then per relation r: 1+4r:row, 2+4r:col, 3+4r:W, 4+4r:a,
// 13:bias. Output: [N,128] fp32.
// ---------------------------------------------------------------------------
extern "C" void kernel_launch(void* const* d_in, const int* in_sizes, int n_in,
                              void* d_out, int out_size, void* d_ws, size_t ws_size,
                              hipStream_t stream) {
    const float* H    = (const float*)d_in[0];
    const float* bias = (const float*)d_in[13];
    const int nNodes  = in_sizes[0] / D;
    float* out = (float*)d_out;

    auto align256 = [](size_t x) { return (x + 255) & ~(size_t)255; };
    char* ws = (char*)d_ws;
    float* Wh    = (float*)ws; ws += align256((size_t)nNodes * D * sizeof(float));
    float* s_dst = (float*)ws; ws += align256((size_t)nNodes * sizeof(float));
    float* s_src = (float*)ws; ws += align256((size_t)nNodes * sizeof(float));
    float* e_max = (float*)ws; ws += align256((size_t)nNodes * sizeof(float));
    float* e_sum = (float*)ws; ws += align256((size_t)nNodes * sizeof(float));
    float* e_buf = (float*)ws;  // sized by largest E

    // out = bias (broadcast)
    {
        long total4 = (long)nNodes * (D / 4);
        init_out_kernel<<<(int)((total4 + 255) / 256), 256, 0, stream>>>(out, bias, total4);
    }

    for (int r = 0; r < 3; ++r) {
        const int*   row = (const int*)d_in[1 + 4 * r];
        const int*   col = (const int*)d_in[2 + 4 * r];
        const float* W   = (const float*)d_in[3 + 4 * r];
        const float* a   = (const float*)d_in[4 + 4 * r];
        const int    E   = in_sizes[1 + 4 * r];

        clear_kernel<<<(nNodes + 255) / 256, 256, 0, stream>>>(e_max, e_sum, nNodes);
        gat_gemm_kernel<<<(nNodes + 63) / 64, 128, 0, stream>>>(H, W, Wh, nNodes);
        score_kernel<<<(nNodes + 7) / 8, 256, 0, stream>>>(Wh, a, s_dst, s_src, nNodes);
        edge_max_kernel<<<(E + 255) / 256, 256, 0, stream>>>(row, col, s_dst, s_src,
                                                             e_buf, e_max, E);
        edge_exp_kernel<<<(E + 255) / 256, 256, 0, stream>>>(row, e_buf, e_max, e_sum, E);
        agg_kernel<<<(E + 7) / 8, 256, 0, stream>>>(row, col, e_buf, e_sum, Wh, out, E);
    }
}